// SAGE_7851200217408
// MI455X (gfx1250) — compile-verified
//
#include <hip/hip_runtime.h>
#include <math.h>

#define HD 128           // feature width D == H == 128
#define BN_EPS 1e-5f

typedef __attribute__((ext_vector_type(2))) float v2f;
typedef __attribute__((ext_vector_type(8))) float v8f;

// ---------------------------------------------------------------------------
// Utility: zero a float buffer (grid-stride)
// ---------------------------------------------------------------------------
__global__ void zero_f32(float* __restrict__ p, size_t n) {
  size_t i = (size_t)blockIdx.x * blockDim.x + threadIdx.x;
  size_t step = (size_t)gridDim.x * blockDim.x;
  for (; i < n; i += step) p[i] = 0.0f;
}

// ---------------------------------------------------------------------------
// Degree: one thread per edge, atomic add 1.0 at dst
// ---------------------------------------------------------------------------
__global__ void deg_kernel(const int* __restrict__ dst, float* __restrict__ deg, int E) {
  int e = blockIdx.x * blockDim.x + threadIdx.x;
  if (e < E) atomicAdd(&deg[dst[e]], 1.0f);
}

// ---------------------------------------------------------------------------
// Scatter: one wave (32 lanes) per edge; lane l handles features [4l, 4l+4)
// float4 coalesced gather of x[src], per-element atomic add into agg[dst].
// ---------------------------------------------------------------------------
__global__ void scatter_kernel(const float* __restrict__ x,
                               const int* __restrict__ dstA,
                               const int* __restrict__ srcA,
                               float* __restrict__ agg, int E) {
  long gid = (long)blockIdx.x * blockDim.x + threadIdx.x;
  int e = (int)(gid >> 5);
  int lane = (int)(gid & 31);
  if (e >= E) return;
  int d = dstA[e];
  int s = srcA[e];
  const float4 v = ((const float4*)(x + (size_t)s * HD))[lane];
  float* ap = agg + (size_t)d * HD + lane * 4;
  atomicAdd(ap + 0, v.x);
  atomicAdd(ap + 1, v.y);
  atomicAdd(ap + 2, v.z);
  atomicAdd(ap + 3, v.w);
}

// ---------------------------------------------------------------------------
// Mean: agg[i,:] *= 1 / max(deg[i], 1)
// ---------------------------------------------------------------------------
__global__ void scale_by_deg(float* __restrict__ agg, const float* __restrict__ deg, int N) {
  long gid = (long)blockIdx.x * blockDim.x + threadIdx.x;
  int row = (int)(gid >> 5);
  int j = (int)(gid & 31);
  if (row >= N) return;
  float inv = 1.0f / fmaxf(deg[row], 1.0f);
  float4* p = (float4*)(agg + (size_t)row * HD) + j;
  float4 v = *p;
  v.x *= inv; v.y *= inv; v.z *= inv; v.w *= inv;
  *p = v;
}

// ---------------------------------------------------------------------------
// Fused dual GEMM via V_WMMA_F32_16X16X4_F32 (exact fp32 matrix path):
//   out = A1 @ W1^T + A2 @ W2^T + bias          (A: [M,128], W: [NOUT,128])
// Block = 256 threads = 8 waves; each wave owns a 32-row M tile (two 16-row
// WMMA tiles sharing B fragments -> each ds_load_b64 feeds 2 WMMAs) and all
// NT = ceil(NOUT/16) column tiles. Weights staged in LDS, row pitch 132
// (16B-aligned float4 fills; v2f fragment reads are bank-conflict-free).
// Out-of-range M rows are CLAMPED (not branched): garbage rows never stored.
//
// ISA fragment layouts (CDNA5 ISA):
//   A 16x4 f32 : lane<16 -> M=lane, K=k0+{0,1}; lane>=16 -> M=lane-16, K=k0+{2,3}
//   B  4x16 f32: lane<16 -> N=lane, K=k0+{0,1}; lane>=16 -> N=lane-16, K=k0+{2,3}
//   C/D 16x16  : VGPR r  -> M=r (lanes 0-15) / M=r+8 (lanes 16-31), N=lane&15
// ---------------------------------------------------------------------------
template <int NOUT>
__global__ void __launch_bounds__(256) sage_gemm_kernel(
    const float* __restrict__ A1, const float* __restrict__ W1,
    const float* __restrict__ A2, const float* __restrict__ W2,
    const float* __restrict__ bias, float* __restrict__ out, int M) {
  constexpr int NT = (NOUT + 15) / 16;
  constexpr int NPAD = NT * 16;
  constexpr int LDW = HD + 4;  // 132: keeps float4 alignment + spreads LDS banks
  extern __shared__ float smem[];
  float* lw1 = smem;
  float* lw2 = smem + NPAD * LDW;

  const int tid = threadIdx.x;

  // Cooperative staging of both weight matrices into LDS (zero-pad rows >= NOUT)
  for (int i = tid; i < NPAD * (HD / 4); i += 256) {
    int r = i / (HD / 4);
    int c4 = (i % (HD / 4)) * 4;
    float4 w1 = make_float4(0.f, 0.f, 0.f, 0.f);
    float4 w2 = w1;
    if (r < NOUT) {
      w1 = *(const float4*)(W1 + (size_t)r * HD + c4);
      w2 = *(const float4*)(W2 + (size_t)r * HD + c4);
    }
    *(float4*)(lw1 + r * LDW + c4) = w1;
    *(float4*)(lw2 + r * LDW + c4) = w2;
  }
  __syncthreads();

  const int wave = tid >> 5;
  const int lane = tid & 31;
  const int half = lane >> 4;  // selects K sub-pair {0,1} vs {2,3}
  const int l16 = lane & 15;
  const int mbase = blockIdx.x * 256 + wave * 32;  // 32 rows per wave

  // Clamp out-of-range rows instead of branching: rows >= M compute garbage
  // that is masked at the store, and clamping keeps EXEC uniform for WMMA.
  const int r0 = min(mbase + l16, M - 1);
  const int r1 = min(mbase + 16 + l16, M - 1);
  const float* a1p0 = A1 + (size_t)r0 * HD + half * 2;
  const float* a1p1 = A1 + (size_t)r1 * HD + half * 2;
  const float* a2p0 = A2 + (size_t)r0 * HD + half * 2;
  const float* a2p1 = A2 + (size_t)r1 * HD + half * 2;

  v8f acc[2][NT] = {};

  // First product: A1 @ W1^T
#pragma unroll 4
  for (int k0 = 0; k0 < HD; k0 += 4) {
    v2f a0 = *(const v2f*)(a1p0 + k0);
    v2f a1 = *(const v2f*)(a1p1 + k0);
#pragma unroll
    for (int t = 0; t < NT; ++t) {
      v2f b = *(const v2f*)(lw1 + (t * 16 + l16) * LDW + k0 + half * 2);
      acc[0][t] = __builtin_amdgcn_wmma_f32_16x16x4_f32(
          false, a0, false, b, (short)0, acc[0][t], false, false);
      acc[1][t] = __builtin_amdgcn_wmma_f32_16x16x4_f32(
          false, a1, false, b, (short)0, acc[1][t], false, false);
    }
  }
  // Second product accumulated on top: A2 @ W2^T
#pragma unroll 4
  for (int k0 = 0; k0 < HD; k0 += 4) {
    v2f a0 = *(const v2f*)(a2p0 + k0);
    v2f a1 = *(const v2f*)(a2p1 + k0);
#pragma unroll
    for (int t = 0; t < NT; ++t) {
      v2f b = *(const v2f*)(lw2 + (t * 16 + l16) * LDW + k0 + half * 2);
      acc[0][t] = __builtin_amdgcn_wmma_f32_16x16x4_f32(
          false, a0, false, b, (short)0, acc[0][t], false, false);
      acc[1][t] = __builtin_amdgcn_wmma_f32_16x16x4_f32(
          false, a1, false, b, (short)0, acc[1][t], false, false);
    }
  }

  // Bias + store (C/D layout): VGPR r -> row base + r + 8*half, col t*16 + l16
#pragma unroll
  for (int s = 0; s < 2; ++s) {
#pragma unroll
    for (int t = 0; t < NT; ++t) {
      int col = t * 16 + l16;
      float bv = (col < NOUT) ? bias[col] : 0.f;
#pragma unroll
      for (int r = 0; r < 8; ++r) {
        int row = mbase + s * 16 + r + half * 8;
        if (row < M && col < NOUT)
          out[(size_t)row * NOUT + col] = acc[s][t][r] + bv;
      }
    }
  }
}

// ---------------------------------------------------------------------------
// BatchNorm statistics: per-channel sum and sum-of-squares over N rows.
// 256 threads = channels x 2 row-groups; block-partial reduce then atomics.
// ---------------------------------------------------------------------------
__global__ void bn_stats(const float* __restrict__ x, float* __restrict__ sums, int N) {
  __shared__ float ls[2][256];
  int c = threadIdx.x & (HD - 1);
  int halfg = threadIdx.x >> 7;
  float s = 0.f, s2 = 0.f;
  for (int row = blockIdx.x * 2 + halfg; row < N; row += gridDim.x * 2) {
    float v = x[(size_t)row * HD + c];
    s += v;
    s2 += v * v;
  }
  ls[0][threadIdx.x] = s;
  ls[1][threadIdx.x] = s2;
  __syncthreads();
  if (halfg == 0) {
    s += ls[0][c + HD];
    s2 += ls[1][c + HD];
    atomicAdd(&sums[c], s);
    atomicAdd(&sums[HD + c], s2);
  }
}

// scale = g * rsqrt(var + eps); shift = b - mu * scale   (biased variance)
__global__ void bn_finalize(const float* __restrict__ sums,
                            const float* __restrict__ g,
                            const float* __restrict__ b,
                            float* __restrict__ sc, int N) {
  int c = threadIdx.x;
  if (c >= HD) return;
  float invn = 1.0f / (float)N;
  float mu = sums[c] * invn;
  float var = fmaxf(sums[HD + c] * invn - mu * mu, 0.0f);
  float s = g[c] * rsqrtf(var + BN_EPS);
  sc[c] = s;
  sc[HD + c] = b[c] - mu * s;
}

__global__ void bn_apply_relu(float* __restrict__ x, const float* __restrict__ sc, long total) {
  long i = (long)blockIdx.x * blockDim.x + threadIdx.x;
  if (i >= total) return;
  int c = (int)(i & (HD - 1));
  float v = x[i] * sc[c] + sc[HD + c];
  x[i] = fmaxf(v, 0.0f);
}

// ---------------------------------------------------------------------------
// Host orchestration
// ---------------------------------------------------------------------------
extern "C" void kernel_launch(void* const* d_in, const int* in_sizes, int n_in,
                              void* d_out, int out_size, void* d_ws, size_t ws_size,
                              hipStream_t stream) {
  const float* x   = (const float*)d_in[0];
  const int*   ei  = (const int*)d_in[1];
  const float* Wl0 = (const float*)d_in[2];
  const float* bl0 = (const float*)d_in[3];
  const float* Wr0 = (const float*)d_in[4];
  const float* g0  = (const float*)d_in[5];
  const float* b0  = (const float*)d_in[6];
  const float* Wl1 = (const float*)d_in[7];
  const float* bl1 = (const float*)d_in[8];
  const float* Wr1 = (const float*)d_in[9];
  const float* g1  = (const float*)d_in[10];
  const float* b1  = (const float*)d_in[11];
  const float* Wl2 = (const float*)d_in[12];
  const float* bl2 = (const float*)d_in[13];
  const float* Wr2 = (const float*)d_in[14];

  const int N = in_sizes[0] / HD;
  const int E = in_sizes[1] / 2;

  // Workspace carve-up (floats)
  float* agg   = (float*)d_ws;
  float* h0    = agg + (size_t)N * HD;
  float* h1    = h0 + (size_t)N * HD;
  float* deg   = h1 + (size_t)N * HD;
  float* bnsum = deg + N;          // 2*HD
  float* bnsc  = bnsum + 2 * HD;   // 2*HD

  const int* dstA = ei;
  const int* srcA = ei + E;

  const int scatterBlocks = (E * 32 + 255) / 256;
  const int scaleBlocks   = (N * 32 + 255) / 256;
  const int gemmBlocks    = (N + 255) / 256;
  const long totalNH      = (long)N * HD;
  const int applyBlocks   = (int)((totalNH + 255) / 256);
  const size_t lds128 = (size_t)2 * 128 * (HD + 4) * sizeof(float);  // 132 KB
  const size_t lds48  = (size_t)2 * 48  * (HD + 4) * sizeof(float);

  // Degree (shared by all layers)
  zero_f32<<<1024, 256, 0, stream>>>(deg, (size_t)N);
  deg_kernel<<<(E + 255) / 256, 256, 0, stream>>>(dstA, deg, E);

  // ---- Layer 0 ----
  zero_f32<<<4096, 256, 0, stream>>>(agg, (size_t)N * HD);
  scatter_kernel<<<scatterBlocks, 256, 0, stream>>>(x, dstA, srcA, agg, E);
  scale_by_deg<<<scaleBlocks, 256, 0, stream>>>(agg, deg, N);
  sage_gemm_kernel<128><<<gemmBlocks, 256, lds128, stream>>>(agg, Wl0, x, Wr0, bl0, h0, N);
  zero_f32<<<1, 256, 0, stream>>>(bnsum, (size_t)(2 * HD));
  bn_stats<<<512, 256, 0, stream>>>(h0, bnsum, N);
  bn_finalize<<<1, HD, 0, stream>>>(bnsum, g0, b0, bnsc, N);
  bn_apply_relu<<<applyBlocks, 256, 0, stream>>>(h0, bnsc, totalNH);

  // ---- Layer 1 ----
  zero_f32<<<4096, 256, 0, stream>>>(agg, (size_t)N * HD);
  scatter_kernel<<<scatterBlocks, 256, 0, stream>>>(h0, dstA, srcA, agg, E);
  scale_by_deg<<<scaleBlocks, 256, 0, stream>>>(agg, deg, N);
  sage_gemm_kernel<128><<<gemmBlocks, 256, lds128, stream>>>(agg, Wl1, h0, Wr1, bl1, h1, N);
  zero_f32<<<1, 256, 0, stream>>>(bnsum, (size_t)(2 * HD));
  bn_stats<<<512, 256, 0, stream>>>(h1, bnsum, N);
  bn_finalize<<<1, HD, 0, stream>>>(bnsum, g1, b1, bnsc, N);
  bn_apply_relu<<<applyBlocks, 256, 0, stream>>>(h1, bnsc, totalNH);

  // ---- Layer 2 (output, C = 47 channels) ----
  zero_f32<<<4096, 256, 0, stream>>>(agg, (size_t)N * HD);
  scatter_kernel<<<scatterBlocks, 256, 0, stream>>>(h1, dstA, srcA, agg, E);
  scale_by_deg<<<scaleBlocks, 256, 0, stream>>>(agg, deg, N);
  sage_gemm_kernel<47><<<gemmBlocks, 256, lds48, stream>>>(agg, Wl2, h1, Wr2, bl2,
                                                           (float*)d_out, N);
}